// CombinedEmbedding_20761871909648
// MI455X (gfx1250) — compile-verified
//
#include <hip/hip_runtime.h>
#include <math.h>

#define NB 32
#define NS 2048
#define NT (NB * NS)
#define NF 8
#define NE 8
#define D_NODE 128
#define D_BT 32
#define D_POP 16
#define D_ET 32
#define D_EQ 32
#define D_PROJ 304
#define D_OUT 64
#define TILE_M 64
#define KSTRIDE 306   // 304 + 2 pad: even (b64 align), spreads LDS banks

typedef float v2f __attribute__((ext_vector_type(2)));
typedef float v8f __attribute__((ext_vector_type(8)));

__global__ void counts_sum_kernel(const float* __restrict__ counts,
                                  float* __restrict__ csum) {
    int s = blockIdx.x * blockDim.x + threadIdx.x;
    if (s < NS) {
        float acc = 0.0f;
        for (int b = 0; b < NB; ++b) acc += counts[b * NS + s];
        csum[s] = acc;
    }
}

__global__ __launch_bounds__(256, 1)
void fused_embed_proj_kernel(const int*   __restrict__ nid,
                             const float* __restrict__ tf,
                             const int*   __restrict__ btid,
                             const float* __restrict__ pop,
                             const int*   __restrict__ etid,
                             const int*   __restrict__ eqid,
                             const float* __restrict__ node_tab,
                             const float* __restrict__ t2v_w,
                             const float* __restrict__ t2v_b,
                             const float* __restrict__ bt_tab,
                             const float* __restrict__ pop_w,
                             const float* __restrict__ pop_b,
                             const float* __restrict__ et_tab,
                             const float* __restrict__ eq_tab,
                             const float* __restrict__ proj_w,
                             const float* __restrict__ proj_b,
                             const float* __restrict__ csum,
                             float*       __restrict__ out) {
    __shared__ __align__(16) float sA[TILE_M * KSTRIDE];   // combined tile, row-major [token][k]
    __shared__ __align__(16) float sW[D_OUT * KSTRIDE];    // proj_w transposed [n][k]

    const int tid = threadIdx.x;
    const int t0  = blockIdx.x * TILE_M;

    // ---- stage proj_w transposed: sW[n][k] = proj_w[k*64 + n] (coalesced global read)
    for (int i = tid; i < D_PROJ * D_OUT; i += 256) {
        int k = i >> 6;
        int n = i & 63;
        sW[n * KSTRIDE + k] = proj_w[i];
    }

    // ---- spatial cols [0,128): gather node2vec rows (128 consecutive lanes per row)
    for (int i = tid; i < TILE_M * D_NODE; i += 256) {
        int r = i >> 7;
        int c = i & 127;
        sA[r * KSTRIDE + c] = node_tab[(size_t)nid[t0 + r] * D_NODE + c];
    }

    // ---- Time2Vec cols [128,192): per-feature affine, e==0 raw else sin
    for (int i = tid; i < TILE_M * (NF * NE); i += 256) {
        int r = i >> 6;
        int j = i & 63;
        int f = j >> 3;
        int e = j & 7;
        float x  = tf[(t0 + r) * NF + f];
        float xa = fmaf(x, t2v_w[f * NE + e], t2v_b[f * NE + e]);
        sA[r * KSTRIDE + 128 + j] = (e == 0) ? xa : sinf(xa);
    }

    // ---- cols [192,304): building*countsum | pop linear | etype | equip
    for (int i = tid; i < TILE_M * 112; i += 256) {
        int r = i / 112;
        int c = i % 112;
        int t = t0 + r;
        float v;
        if (c < 32) {
            v = bt_tab[btid[t] * D_BT + c] * csum[t & (NS - 1)];
        } else if (c < 48) {
            int cc = c - 32;
            v = fmaf(pop[t], pop_w[cc], pop_b[cc]);
        } else if (c < 80) {
            v = et_tab[etid[t] * D_ET + (c - 48)];
        } else {
            v = eq_tab[eqid[t] * D_EQ + (c - 80)];
        }
        sA[r * KSTRIDE + 192 + c] = v;
    }

    __syncthreads();

    // ---- WMMA projection: 4 M-tiles x 4 N-tiles; wave w -> mt = w&3, nt = {2*(w>>2), +1}
    const int wave = tid >> 5;
    const int lane = tid & 31;
    const int mt   = wave & 3;
    const int nt0  = (wave >> 2) * 2;
    const int lm   = lane & 15;          // M for A-frag / N for B-frag
    const int kh   = (lane >> 4) * 2;    // lanes 16-31 hold K+2

    v8f c0 = {0.f,0.f,0.f,0.f,0.f,0.f,0.f,0.f};
    v8f c1 = c0;

    const float* aRow  = &sA[(mt * 16 + lm) * KSTRIDE + kh];
    const float* bRow0 = &sW[(nt0 * 16 + lm) * KSTRIDE + kh];
    const float* bRow1 = &sW[((nt0 + 1) * 16 + lm) * KSTRIDE + kh];

    #pragma unroll 4
    for (int ks = 0; ks < D_PROJ; ks += 4) {
        v2f a  = *(const v2f*)(aRow  + ks);   // {A[M][k+kh], A[M][k+kh+1]}
        v2f b0 = *(const v2f*)(bRow0 + ks);   // {W[k+kh][n], W[k+kh+1][n]}
        v2f b1 = *(const v2f*)(bRow1 + ks);
        c0 = __builtin_amdgcn_wmma_f32_16x16x4_f32(false, a, false, b0,
                                                   (short)0, c0, false, false);
        c1 = __builtin_amdgcn_wmma_f32_16x16x4_f32(false, a, false, b1,
                                                   (short)0, c1, false, false);
    }

    // ---- store D + bias; C/D layout: VGPR v, lane l -> M = v + 8*(l>>4), N = l&15
    const int n0 = nt0 * 16 + lm;
    const int n1 = (nt0 + 1) * 16 + lm;
    const float bias0 = proj_b[n0];
    const float bias1 = proj_b[n1];
    const int mbase = t0 + mt * 16 + ((lane >> 4) << 3);
    #pragma unroll
    for (int v = 0; v < 8; ++v) {
        out[(size_t)(mbase + v) * D_OUT + n0] = c0[v] + bias0;
        out[(size_t)(mbase + v) * D_OUT + n1] = c1[v] + bias1;
    }
}

extern "C" void kernel_launch(void* const* d_in, const int* in_sizes, int n_in,
                              void* d_out, int out_size, void* d_ws, size_t ws_size,
                              hipStream_t stream) {
    const int*   nid    = (const int*)  d_in[0];
    const float* tf     = (const float*)d_in[1];
    const int*   btid   = (const int*)  d_in[2];
    const float* counts = (const float*)d_in[3];
    const float* pop    = (const float*)d_in[4];
    const int*   etid   = (const int*)  d_in[5];
    const int*   eqid   = (const int*)  d_in[6];
    const float* ntab   = (const float*)d_in[7];
    const float* t2vw   = (const float*)d_in[8];
    const float* t2vb   = (const float*)d_in[9];
    const float* bttab  = (const float*)d_in[10];
    const float* popw   = (const float*)d_in[11];
    const float* popb   = (const float*)d_in[12];
    const float* ettab  = (const float*)d_in[13];
    const float* eqtab  = (const float*)d_in[14];
    const float* projw  = (const float*)d_in[15];
    const float* projb  = (const float*)d_in[16];
    float* out  = (float*)d_out;
    float* csum = (float*)d_ws;   // NS floats

    counts_sum_kernel<<<(NS + 255) / 256, 256, 0, stream>>>(counts, csum);

    fused_embed_proj_kernel<<<NT / TILE_M, 256, 0, stream>>>(
        nid, tf, btid, pop, etid, eqid,
        ntab, t2vw, t2vb, bttab, popw, popb, ettab, eqtab,
        projw, projb, csum, out);
}